// MyLoss_53403623358552
// MI455X (gfx1250) — compile-verified
//
#include <hip/hip_runtime.h>
#include <cstdint>

// ---------------------------------------------------------------------------
// Trajectory dead-reckoning + haversine loss for MI455X (gfx1250, wave32).
//
// Layouts (float32):
//   pred: (B, 256, 2)   row pitch 2048 B
//   true: (B, 257, 4)   row pitch 4112 B
//   out : [loss, posi_preds(B,256,2) flat]  (out_size = 1 + B*512)
//
// 1 wave (32 lanes) per block, one batch row per lane, sequential 256-step
// scan in registers. Per-tile (8 steps) staging of pred + true lon/lat into
// LDS via global_load_async_to_lds_b64 (coalesced 64 B row segments), double
// buffered with partial s_wait_asynccnt — wave-local, so no barriers needed.
// Output stores are non-temporal (write-only stream; keep L2 for reads).
// ---------------------------------------------------------------------------

#define DEG2RADf 0.017453292519943295f
#define RAD2DEGf 57.29577951308232f

#define TT     8     // timesteps per tile
#define NT     32    // 256 / TT
#define PITCHF 18    // floats per LDS row (16 data + 2 pad = 72 B, multiple of 8)

__device__ __forceinline__ unsigned lds_off(const void* p) {
    // Generic pointer to LDS = {shared aperture, 32-bit LDS offset}; the low
    // 32 bits are the wave-relative LDS address the async engine consumes.
    return (unsigned)(uintptr_t)p;
}

__device__ __forceinline__ void async_b64(const void* g, unsigned lds) {
    asm volatile("global_load_async_to_lds_b64 %0, %1, off"
                 :: "v"(lds), "v"((unsigned long long)(uintptr_t)g)
                 : "memory");
}

#define WAIT_ASYNC(n) asm volatile("s_wait_asynccnt " #n ::: "memory")

__device__ __forceinline__ float earth_radius_km(float s, float c) {
    const float Af = 6378137.0f, Bf = 6356752.3142f;
    const float A2 = Af * Af,     B2 = Bf * Bf;
    float an = A2 * c, bn = B2 * s;
    float ad = Af * c, bd = Bf * s;
    return sqrtf((an * an + bn * bn) / (ad * ad + bd * bd) + 1e-10f) * 0.001f;
}

__device__ __forceinline__ void stage_tile(const char* predB, const char* trueB,
                                           int b0, int t0, float* sP, float* sT,
                                           int tid) {
    // pred: 32 rows x 8 chunks of 8 B (one chunk = one step's {dsog,dcog});
    // lanes 0..7 cover one row's 64 contiguous bytes -> coalesced.
#pragma unroll
    for (int j = 0; j < 8; ++j) {
        int idx = j * 32 + tid;          // 0..255
        int r = idx >> 3, c = idx & 7;
        const char* g = predB + (size_t)(b0 + r) * 2048u + (size_t)t0 * 8u + (size_t)c * 8u;
        async_b64(g, lds_off(sP + r * PITCHF + c * 2));
    }
    // true: 32 rows x 8 records; only first 8 B (lon,lat) of each 16 B record
#pragma unroll
    for (int j = 0; j < 8; ++j) {
        int idx = j * 32 + tid;
        int r = idx >> 3, c = idx & 7;
        const char* g = trueB + (size_t)(b0 + r) * 4112u + (size_t)(t0 + 1 + c) * 16u;
        async_b64(g, lds_off(sT + r * PITCHF + c * 2));
    }
}

__global__ __launch_bounds__(32) void traj_kernel(const float* __restrict__ pred,
                                                  const float* __restrict__ truep,
                                                  float* __restrict__ out,
                                                  float* __restrict__ partial) {
    __shared__ __align__(16) float sPred[2][32 * PITCHF];
    __shared__ __align__(16) float sTrue[2][32 * PITCHF];

    const int tid = threadIdx.x;           // lane = row within wave
    const int b0  = blockIdx.x * 32;
    const int b   = b0 + tid;

    const char* predB = (const char*)pred;
    const char* trueB = (const char*)truep;

    // ---- initial state from true[b][0] = (lon, lat, sog, cog) -------------
    float4 t0v = *(const float4*)(truep + (size_t)b * 1028u);
    float sog = t0v.z;
    float cog = t0v.w;
    float lonr = t0v.x * DEG2RADf;
    float latr = t0v.y * DEG2RADf;
    float sinlat, coslat;
    sincosf(latr, &sinlat, &coslat);
    float radius = earth_radius_km(sinlat, coslat);

    // ---- prologue: stage tile 0 ------------------------------------------
    stage_tile(predB, trueB, b0, 0, sPred[0], sTrue[0], tid);

    const size_t outbase = 1u + (size_t)b * 512u;
    float lsum = 0.0f;

    for (int it = 0; it < NT; ++it) {
        int cbuf = it & 1;
        if (it + 1 < NT) {
            stage_tile(predB, trueB, b0, (it + 1) * TT,
                       sPred[(it + 1) & 1], sTrue[(it + 1) & 1], tid);
            WAIT_ASYNC(16);   // tile `it` done; next tile's 16 ops may remain
        } else {
            WAIT_ASYNC(0);
        }

        const float* myP = sPred[cbuf] + tid * PITCHF;
        const float* myT = sTrue[cbuf] + tid * PITCHF;

        // unroll-by-2: enough to overlap step t's haversine with step t+1's
        // advance (carried chain is short), without blowing out the I$.
#pragma unroll 2
        for (int s = 0; s < TT; ++s) {
            float dsg = myP[2 * s];
            float dcg = myP[2 * s + 1];
            sog += dsg;
            cog += dcg;
            cog = (cog > 360.0f) ? (cog - 360.0f)
                                 : ((cog < 0.0f) ? (cog + 360.0f) : cog);
            float cogr = cog * DEG2RADf;

            // d <= ~6e-4 rad: 2-term series is below fp32 ulp of sinf/cosf
            float d  = (sog * (3.0f / 60.0f)) * 1.852f / radius;
            float d2 = d * d;
            float sind = d * (1.0f - 0.16666667f * d2);
            float cosd = 1.0f - 0.5f * d2;

            float sc, cc;
            sincosf(cogr, &sc, &cc);

            float slat2 = sinlat * cosd + coslat * sind * cc;
            float lat2r = asinf(slat2);
            float y = sc * sind * coslat;
            float x = cosd - sinlat * slat2;
            float lon2r = lonr + atan2f(y, x);

            float lon2d = lon2r * RAD2DEGf;
            float lat2d = lat2r * RAD2DEGf;

            int t = it * TT + s;
            __builtin_nontemporal_store(lon2d, &out[outbase + 2 * (size_t)t]);
            __builtin_nontemporal_store(lat2d, &out[outbase + 2 * (size_t)t + 1]);

            // carry sin/cos(lat) and radius (radius(lat_t) is reused by both
            // this step's haversine and the next step's advance)
            sinlat = slat2;
            coslat = sqrtf(fmaxf(1.0f - slat2 * slat2, 0.0f));
            radius = earth_radius_km(sinlat, coslat);

            // haversine vs true[b][t+1]
            float lonT = myT[2 * s];
            float latT = myT[2 * s + 1];
            float sdlat = sinf((latT - lat2d) * (0.5f * DEG2RADf));
            float sdlon = sinf((lonT - lon2d) * (0.5f * DEG2RADf));
            float a = sdlat * sdlat + coslat * cosf(latT * DEG2RADf) * sdlon * sdlon;
            float c2 = 2.0f * atan2f(sqrtf(a + 1e-10f), sqrtf(1.0f - a + 1e-10f));
            lsum += radius * c2;

            lonr = lon2r;
        }
    }

    // deterministic wave32 butterfly reduction
#pragma unroll
    for (int o = 16; o > 0; o >>= 1) lsum += __shfl_xor(lsum, o, 32);
    if (tid == 0) partial[blockIdx.x] = lsum;
}

__global__ __launch_bounds__(256) void reduce_kernel(const float* __restrict__ partial,
                                                     float* __restrict__ out,
                                                     int n, float scale) {
    __shared__ float sh[256];
    int t = threadIdx.x;
    float v = 0.0f;
    for (int i = t; i < n; i += 256) v += partial[i];
    sh[t] = v;
    __syncthreads();
    for (int o = 128; o > 0; o >>= 1) {
        if (t < o) sh[t] += sh[t + o];
        __syncthreads();
    }
    if (t == 0) out[0] = sh[0] * scale;
}

extern "C" void kernel_launch(void* const* d_in, const int* in_sizes, int n_in,
                              void* d_out, int out_size, void* d_ws, size_t ws_size,
                              hipStream_t stream) {
    const float* pred  = (const float*)d_in[0];   // (B, 256, 2)
    const float* truep = (const float*)d_in[1];   // (B, 257, 4)
    float* out     = (float*)d_out;
    float* partial = (float*)d_ws;

    int B = in_sizes[1] / (257 * 4);              // 32768
    int nblocks = B / 32;                          // 1024 single-wave blocks

    traj_kernel<<<nblocks, 32, 0, stream>>>(pred, truep, out, partial);
    reduce_kernel<<<1, 256, 0, stream>>>(partial, out, nblocks,
                                         1.0f / ((float)B * 256.0f));
}